// VarDecoderGru_14559939134103
// MI455X (gfx1250) — compile-verified
//
#include <hip/hip_runtime.h>

// Autoregressive GRU decoder for MI455X (gfx1250), fp32 WMMA path.
// Sequence per step t: k_gates (WMMA) -> k_gru (elementwise) ->
// k_logits (WMMA, LDS-staged A) -> k_topk (wave32 shuffle reduction).

#define VOCAB 32000
#define DIM   512
#define HID   512
#define BATCH 32
#define TLEN  128
#define NTOPK 5
#define BOS   1
#define ALPHA 0.2f
#define G3    (3*HID)   // 1536

typedef __attribute__((ext_vector_type(2))) float v2f;
typedef __attribute__((ext_vector_type(8))) float v8f;

__device__ __forceinline__ v8f wmma_f32(v2f a, v2f b, v8f c) {
  // D = A(16x4 f32) x B(4x16 f32) + C(16x16 f32)
  return __builtin_amdgcn_wmma_f32_16x16x4_f32(
      /*neg_a=*/false, a, /*neg_b=*/false, b,
      /*c_mod=*/(short)0, c, /*reuse_a=*/false, /*reuse_b=*/false);
}

// ---------------------------------------------------------------- init
__global__ void k_init(const float* __restrict__ ctx, const float* __restrict__ E,
                       float* __restrict__ h, float* __restrict__ x,
                       int* __restrict__ tok) {
  int i = blockIdx.x * blockDim.x + threadIdx.x;
  if (i < BATCH * HID) {
    h[i] = ctx[i];
    float e = E[BOS * DIM + (i & (DIM - 1))];
    x[i] = e > 0.f ? e : ALPHA * e;
  }
  if (i < BATCH) tok[i] = BOS;
}

// ------------------------------------------------- gate GEMMs (WMMA f32)
// xg = x @ Wx ; hg = h @ U   with x,h: [32,512], Wx,U: [512,1536]
// One wave per 16-column tile (96 waves). Two M-tiles per wave.
__global__ void __launch_bounds__(256) k_gates(
    const float* __restrict__ x, const float* __restrict__ h,
    const float* __restrict__ Wx, const float* __restrict__ U,
    float* __restrict__ xg, float* __restrict__ hg) {
  const int tid   = threadIdx.x;
  const int lane  = tid & 31;
  const int lhalf = lane & 15;
  const int khi   = (lane >> 4) << 1;   // 0 or 2 (K split across half-waves)
  const int rbase = (lane >> 4) << 3;   // 0 or 8 (C row base per half-wave)
  const int wave  = (blockIdx.x * 256 + tid) >> 5;   // 0..95
  const int n     = wave * 16 + lhalf;

  v8f ax0 = {}, ax1 = {}, ah0 = {}, ah1 = {};

  for (int k0 = 0; k0 < DIM; k0 += 4) {
    const int kk = k0 + khi;
    v2f bx, bu, a0x, a1x, a0h, a1h;
    bx[0]  = Wx[(size_t)kk * G3 + n];       bx[1]  = Wx[(size_t)(kk + 1) * G3 + n];
    bu[0]  = U [(size_t)kk * G3 + n];       bu[1]  = U [(size_t)(kk + 1) * G3 + n];
    a0x[0] = x[lhalf * DIM + kk];           a0x[1] = x[lhalf * DIM + kk + 1];
    a1x[0] = x[(lhalf + 16) * DIM + kk];    a1x[1] = x[(lhalf + 16) * DIM + kk + 1];
    a0h[0] = h[lhalf * HID + kk];           a0h[1] = h[(lhalf) * HID + kk + 1];
    a1h[0] = h[(lhalf + 16) * HID + kk];    a1h[1] = h[(lhalf + 16) * HID + kk + 1];
    ax0 = wmma_f32(a0x, bx, ax0);
    ax1 = wmma_f32(a1x, bx, ax1);
    ah0 = wmma_f32(a0h, bu, ah0);
    ah1 = wmma_f32(a1h, bu, ah1);
  }
#pragma unroll
  for (int v = 0; v < 8; ++v) {
    int r = rbase + v;
    xg[(size_t)r * G3 + n]        = ax0[v];
    xg[(size_t)(r + 16) * G3 + n] = ax1[v];
    hg[(size_t)r * G3 + n]        = ah0[v];
    hg[(size_t)(r + 16) * G3 + n] = ah1[v];
  }
}

// ----------------------------------------------- GRU elementwise update
__global__ void k_gru(const float* __restrict__ xg, const float* __restrict__ hg,
                      const float* __restrict__ b_i, const float* __restrict__ b_r,
                      float* __restrict__ h) {
  int i = blockIdx.x * blockDim.x + threadIdx.x;
  if (i >= BATCH * HID) return;
  int b = i >> 9, j = i & (HID - 1);
  const float* xgb = xg + (size_t)b * G3;
  const float* hgb = hg + (size_t)b * G3;
  float z = xgb[j] + b_i[j] + hgb[j] + b_r[j];
  z = 1.f / (1.f + __expf(-z));
  float r = xgb[j + HID] + b_i[j + HID] + hgb[j + HID] + b_r[j + HID];
  r = 1.f / (1.f + __expf(-r));
  float xh = xgb[j + 2 * HID] + b_i[j + 2 * HID];
  float hh = hgb[j + 2 * HID] + b_r[j + 2 * HID];
  float cand = tanhf(xh + r * hh);
  float hp = h[i];
  h[i] = z * hp + (1.f - z) * cand;
}

// --------------------------------------- logits GEMM (WMMA f32, LDS A)
// logits[:, t, :] = h @ Wo + bo, Wo: [512,32000] (L2-resident, 65.5 MB)
__global__ void __launch_bounds__(256) k_logits(
    const float* __restrict__ h, const float* __restrict__ Wo,
    const float* __restrict__ bo, float* __restrict__ out, int t) {
  __shared__ float sh[BATCH * HID];   // 64 KB: h2 staged once per WG
  const int tid = threadIdx.x;
  for (int i = tid; i < BATCH * HID; i += 256) sh[i] = h[i];
  __syncthreads();

  const int lane  = tid & 31;
  const int lhalf = lane & 15;
  const int khi   = (lane >> 4) << 1;
  const int rbase = (lane >> 4) << 3;
  const int wave  = (blockIdx.x * 256 + tid) >> 5;   // 0..1999
  const int n     = wave * 16 + lhalf;

  v8f c0 = {}, c1 = {};
  for (int k0 = 0; k0 < HID; k0 += 4) {
    const int kk = k0 + khi;
    v2f bw, a0, a1;
    bw[0] = Wo[(size_t)kk * VOCAB + n];      bw[1] = Wo[(size_t)(kk + 1) * VOCAB + n];
    a0[0] = sh[lhalf * HID + kk];            a0[1] = sh[lhalf * HID + kk + 1];
    a1[0] = sh[(lhalf + 16) * HID + kk];     a1[1] = sh[(lhalf + 16) * HID + kk + 1];
    c0 = wmma_f32(a0, bw, c0);
    c1 = wmma_f32(a1, bw, c1);
  }
  const float bias = bo[n];
#pragma unroll
  for (int v = 0; v < 8; ++v) {
    int b0 = rbase + v;
    out[((size_t)(b0 * TLEN + t)) * VOCAB + n] = c0[v] + bias;
    int b1 = b0 + 16;
    out[((size_t)(b1 * TLEN + t)) * VOCAB + n] = c1[v] + bias;
  }
}

// ------------------------------- top-k + token select + next embedding
// One wave32 per batch row; 5 masked max passes, ties -> lower index
// (matches jax.lax.top_k ordering).
__global__ void k_topk(const float* __restrict__ out, const int* __restrict__ rand_idx,
                       const float* __restrict__ E, int* __restrict__ tok,
                       float* __restrict__ x, int t) {
  const int b    = blockIdx.x;
  const int lane = threadIdx.x;   // 32 threads
  const float* row = out + ((size_t)(b * TLEN + t)) * VOCAB;

  int besti[NTOPK];
#pragma unroll
  for (int p = 0; p < NTOPK; ++p) besti[p] = -1;

#pragma unroll
  for (int s = 0; s < NTOPK; ++s) {
    float v = -3.4e38f;
    int vi = VOCAB;
    for (int c = lane; c < VOCAB; c += 32) {
      bool used = false;
#pragma unroll
      for (int p = 0; p < NTOPK; ++p) used |= (p < s) && (c == besti[p]);
      if (used) continue;
      float val = row[c];
      if (val > v || (val == v && c < vi)) { v = val; vi = c; }
    }
#pragma unroll
    for (int off = 16; off > 0; off >>= 1) {
      float ov = __shfl_xor(v, off, 32);
      int   oi = __shfl_xor(vi, off, 32);
      if (ov > v || (ov == v && oi < vi)) { v = ov; vi = oi; }
    }
    besti[s] = vi;   // wave-uniform
  }

  const int ridx = rand_idx[t * BATCH + b];
  int sel = besti[0];
#pragma unroll
  for (int p = 0; p < NTOPK; ++p) if (p == ridx) sel = besti[p];
  if (lane == 0) tok[b] = sel;

  // Precompute next step's x = leaky_relu(E[sel])
  for (int d = lane; d < DIM; d += 32) {
    float e = E[(size_t)sel * DIM + d];
    x[b * DIM + d] = e > 0.f ? e : ALPHA * e;
  }
}

// ---------------------------------------------------------------- host
extern "C" void kernel_launch(void* const* d_in, const int* in_sizes, int n_in,
                              void* d_out, int out_size, void* d_ws, size_t ws_size,
                              hipStream_t stream) {
  (void)in_sizes; (void)n_in; (void)out_size; (void)ws_size;
  const float* E   = (const float*)d_in[0];
  const float* Wx  = (const float*)d_in[1];
  const float* U   = (const float*)d_in[2];
  const float* b_i = (const float*)d_in[3];
  const float* b_r = (const float*)d_in[4];
  const float* Wo  = (const float*)d_in[5];
  const float* bo  = (const float*)d_in[6];
  const float* ctx = (const float*)d_in[7];
  const int* rand_idx = (const int*)d_in[8];
  float* out = (float*)d_out;

  float* ws = (float*)d_ws;
  float* h  = ws;             // 16384 f32
  float* x  = ws + 16384;     // 16384 f32
  float* xg = ws + 32768;     // 49152 f32
  float* hg = ws + 81920;     // 49152 f32
  int*  tok = (int*)(ws + 131072);   // 32 i32

  k_init<<<64, 256, 0, stream>>>(ctx, E, h, x, tok);
  for (int t = 0; t < TLEN; ++t) {
    k_gates <<<12,  256, 0, stream>>>(x, h, Wx, U, xg, hg);
    k_gru   <<<64,  256, 0, stream>>>(xg, hg, b_i, b_r, h);
    k_logits<<<250, 256, 0, stream>>>(h, Wo, bo, out, t);
    k_topk  <<<BATCH, 32, 0, stream>>>(out, rand_idx, E, tok, x, t);
  }
}